// Encoder_19542101197188
// MI455X (gfx1250) — compile-verified
//
#include <hip/hip_runtime.h>

#define D 128
#define TILE_M 32
#define EPS 1e-5f

typedef __attribute__((ext_vector_type(2))) float v2f;
typedef __attribute__((ext_vector_type(8))) float v8f;

// ---- CDNA5 async global->LDS copy (ASYNCcnt-tracked), with sync fallback ----
#if __has_builtin(__builtin_amdgcn_global_load_async_to_lds_b128) && \
    __has_builtin(__builtin_amdgcn_s_wait_asynccnt)
#define USE_ASYNC_LDS 1
typedef int v4i_t __attribute__((ext_vector_type(4)));
typedef __attribute__((address_space(1))) v4i_t* gv4i_p;   // global int4*
typedef __attribute__((address_space(3))) v4i_t* lv4i_p;   // LDS int4*
__device__ __forceinline__ void async_copy_b128(const void* g, void* l) {
    // lowers to global_load_async_to_lds_b128 (per-lane 16B, LDS written directly)
    void* gnc = const_cast<void*>(g);
    __builtin_amdgcn_global_load_async_to_lds_b128((gv4i_p)gnc, (lv4i_p)l, 0, 0);
}
__device__ __forceinline__ void wait_async0() {
    __builtin_amdgcn_s_wait_asynccnt(0);
}
#endif

// ---------------- degree / normalization ----------------

__global__ void deg_init_kernel(float* deg, int N) {
    int i = blockIdx.x * blockDim.x + threadIdx.x;
    if (i < N) deg[i] = 1.0f;  // self-loop fill_value = 1
}

__global__ void deg_accum_kernel(const int* __restrict__ dstI,
                                 const float* __restrict__ ew,
                                 float* __restrict__ deg, int E) {
    int e = blockIdx.x * blockDim.x + threadIdx.x;
    if (e < E) atomicAdd(&deg[dstI[e]], ew[e]);
}

__global__ void dinv_kernel(const float* __restrict__ deg,
                            float* __restrict__ dinv, int N) {
    int i = blockIdx.x * blockDim.x + threadIdx.x;
    if (i < N) {
        float d = deg[i];
        dinv[i] = (d > 0.0f) ? rsqrtf(fmaxf(d, 1e-12f)) : 0.0f;
    }
}

// ---------------- GEMM: h = X @ W  (N x 128) @ (128 x 128), fp32 WMMA ----------------
// Block: 256 threads = 8 waves. Block tile = 32 rows x 128 cols.
// wave wm (0..1) picks 16-row half; wave wn (0..3) picks 32-col strip (two 16x16 tiles).
// A tile and K-slabs of W are moved with async global->LDS DMA; W slabs are
// double-buffered so the fetch of slab k+1 overlaps the 8 WMMAs of slab k.

__global__ __launch_bounds__(256)
void gemm_wmma_kernel(const float* __restrict__ X, const float* __restrict__ Wg,
                      float* __restrict__ H, int N) {
    __shared__ float As[TILE_M * 132];   // padded stride 132 -> conflict-free column reads
    __shared__ float Ws[2][16 * D];      // double-buffered 16-row K slabs of W

    const int tid  = threadIdx.x;
    const int lane = tid & 31;
    const int wave = tid >> 5;
    const int wm   = wave >> 2;          // 0..1
    const int wn   = wave & 3;           // 0..3
    const int row0 = blockIdx.x * TILE_M;

#ifdef USE_ASYNC_LDS
    // Issue the whole A tile (32 x 128 f32 = 1024 x b128) asynchronously.
    #pragma unroll
    for (int it = 0; it < 4; ++it) {
        int idx = tid + it * 256;
        int r   = idx >> 5;
        int c4  = idx & 31;
        int grow = row0 + r; if (grow >= N) grow = N - 1;
        async_copy_b128(X + (size_t)grow * D + c4 * 4, As + r * 132 + c4 * 4);
    }
    // Issue W slab 0 (16 x 128 f32 = 512 x b128) asynchronously.
    #pragma unroll
    for (int it = 0; it < 2; ++it) {
        int idx = tid + it * 256;
        int r   = idx >> 5;
        int c4  = idx & 31;
        async_copy_b128(Wg + (size_t)r * D + c4 * 4, &Ws[0][r * D + c4 * 4]);
    }
#else
    // Synchronous A tile staging.
    #pragma unroll
    for (int it = 0; it < 4; ++it) {
        int idx = tid + it * 256;
        int r   = idx >> 5;
        int c4  = idx & 31;
        int grow = row0 + r; if (grow >= N) grow = N - 1;
        float4 v = ((const float4*)(X + (size_t)grow * D))[c4];
        *((float4*)(As + r * 132 + c4 * 4)) = v;
    }
#endif

    v8f acc0 = {};
    v8f acc1 = {};
    const int koff  = (lane < 16) ? 0 : 2;       // A/B: lanes 16-31 hold K+2,K+3 (ISA 7.12.2)
    const int arow  = wm * 16 + (lane & 15);
    const int bcol0 = wn * 32 + (lane & 15);
    const int bcol1 = bcol0 + 16;

    for (int kcI = 0; kcI < 8; ++kcI) {
        const int cur = kcI & 1;
#ifdef USE_ASYNC_LDS
        wait_async0();                   // my async copies for slab kcI (and A) done
        __syncthreads();                 // everyone's done; prior slab fully consumed
        if (kcI < 7) {                   // prefetch slab kcI+1 into the other buffer
            int kc = (kcI + 1) * 16;
            #pragma unroll
            for (int it = 0; it < 2; ++it) {
                int idx = tid + it * 256;
                int r   = idx >> 5;
                int c4  = idx & 31;
                async_copy_b128(Wg + (size_t)(kc + r) * D + c4 * 4,
                                &Ws[cur ^ 1][r * D + c4 * 4]);
            }
        }
#else
        int kc = kcI * 16;
        __syncthreads();                 // protect previous slab's consumers
        #pragma unroll
        for (int it = 0; it < 2; ++it) {
            int idx = tid + it * 256;
            int r   = idx >> 5;
            int c4  = idx & 31;
            float4 v = ((const float4*)(Wg + (size_t)(kc + r) * D))[c4];
            *((float4*)(&Ws[cur][r * D + c4 * 4])) = v;
        }
        __syncthreads();
#endif
        #pragma unroll
        for (int kk = 0; kk < 16; kk += 4) {
            v2f a, b0, b1;
            a.x  = As[arow * 132 + kk + koff];
            a.y  = As[arow * 132 + kk + koff + 1];
            b0.x = Ws[cur][(kk + koff) * D + bcol0];
            b0.y = Ws[cur][(kk + koff + 1) * D + bcol0];
            b1.x = Ws[cur][(kk + koff) * D + bcol1];
            b1.y = Ws[cur][(kk + koff + 1) * D + bcol1];
            acc0 = __builtin_amdgcn_wmma_f32_16x16x4_f32(
                false, a, false, b0, (short)0, acc0, false, false);
            acc1 = __builtin_amdgcn_wmma_f32_16x16x4_f32(
                false, a, false, b1, (short)0, acc1, false, false);
        }
    }

    // C/D layout: VGPR v -> M=v (lanes 0-15), M=v+8 (lanes 16-31); N = lane&15.
    const int colL = lane & 15;
    const int rsel = (lane < 16) ? 0 : 8;
    #pragma unroll
    for (int v = 0; v < 8; ++v) {
        int r = row0 + wm * 16 + rsel + v;
        if (r < N) {
            H[(size_t)r * D + wn * 32 + colL]      = acc0[v];
            H[(size_t)r * D + wn * 32 + 16 + colL] = acc1[v];
        }
    }
}

// ---------------- aggregation ----------------

// agg[n][c] = h[n][c] * dinv[n]^2 (self loop) + bias[c]
__global__ void selfloop_bias_kernel(const float* __restrict__ h,
                                     const float* __restrict__ dinv,
                                     const float* __restrict__ bias,
                                     float* __restrict__ agg, int N) {
    int idx = blockIdx.x * blockDim.x + threadIdx.x;
    if (idx < N * D) {
        int n = idx >> 7;
        int c = idx & (D - 1);
        float di = dinv[n];
        agg[idx] = h[idx] * di * di + bias[c];
    }
}

// One wave per edge; each lane owns a float4 of the 128-wide feature row.
__global__ __launch_bounds__(256)
void edge_scatter_kernel(const int* __restrict__ srcI, const int* __restrict__ dstI,
                         const float* __restrict__ ew, const float* __restrict__ dinv,
                         const float* __restrict__ h, float* __restrict__ agg, int E) {
    int gw   = (blockIdx.x * blockDim.x + threadIdx.x) >> 5;
    int lane = threadIdx.x & 31;
    int wavesTotal = (gridDim.x * blockDim.x) >> 5;

    for (int e = gw; e < E; e += wavesTotal) {
        int s = srcI[e];
        int d = dstI[e];
        float nrm = dinv[s] * ew[e] * dinv[d];

        // prefetch next iteration's source row (lowers to global_prefetch_b8)
        int en = e + wavesTotal;
        if (en < E) {
            __builtin_prefetch(h + (size_t)srcI[en] * D + lane * 4, 0, 0);
        }

        float4 hv = *((const float4*)(h + (size_t)s * D + lane * 4));
        float* op = agg + (size_t)d * D + lane * 4;
        atomicAdd(op + 0, hv.x * nrm);
        atomicAdd(op + 1, hv.y * nrm);
        atomicAdd(op + 2, hv.z * nrm);
        atomicAdd(op + 3, hv.w * nrm);
    }
}

// ---------------- batchnorm ----------------

__global__ void zero_kernel(float* p, int n) {
    int i = blockIdx.x * blockDim.x + threadIdx.x;
    if (i < n) p[i] = 0.0f;
}

// stats[0..127] = sum per channel, stats[128..255] = sum of squares.
__global__ __launch_bounds__(D)
void bn_stats_kernel(const float* __restrict__ agg, float* __restrict__ stats, int N) {
    int c = threadIdx.x;          // channel, coalesced across the block
    float s = 0.0f, sq = 0.0f;
    for (int r = blockIdx.x; r < N; r += gridDim.x) {
        float v = agg[(size_t)r * D + c];
        s += v;
        sq += v * v;
    }
    atomicAdd(&stats[c], s);
    atomicAdd(&stats[D + c], sq);
}

__global__ void bn_apply_kernel(const float* __restrict__ agg,
                                const float* __restrict__ stats,
                                const float* __restrict__ gamma,
                                const float* __restrict__ beta,
                                float* __restrict__ act, int N) {
    int idx = blockIdx.x * blockDim.x + threadIdx.x;
    if (idx < N * D) {
        int c = idx & (D - 1);
        float invN = 1.0f / (float)N;
        float mu   = stats[c] * invN;
        float var  = stats[D + c] * invN - mu * mu;
        float sc   = gamma[c] * rsqrtf(var + EPS);
        float y    = sc * (agg[idx] - mu) + beta[c];
        act[idx]   = fmaxf(y, 0.0f);   // ReLU (dropout p=0)
    }
}

// ---------------- sum pooling ----------------

__global__ __launch_bounds__(256)
void pool_kernel(const float* __restrict__ act, const int* __restrict__ batch,
                 float* __restrict__ out, int N) {
    int gw   = (blockIdx.x * blockDim.x + threadIdx.x) >> 5;
    int lane = threadIdx.x & 31;
    if (gw >= N) return;
    int g = batch[gw];
    float4 v  = *((const float4*)(act + (size_t)gw * D + lane * 4));
    float* op = out + (size_t)g * D + lane * 4;
    atomicAdd(op + 0, v.x);
    atomicAdd(op + 1, v.y);
    atomicAdd(op + 2, v.z);
    atomicAdd(op + 3, v.w);
}

// ---------------- driver ----------------

extern "C" void kernel_launch(void* const* d_in, const int* in_sizes, int n_in,
                              void* d_out, int out_size, void* d_ws, size_t ws_size,
                              hipStream_t stream) {
    const float* x     = (const float*)d_in[0];
    const int*   ei    = (const int*)  d_in[1];
    const float* ew    = (const float*)d_in[2];
    const int*   batch = (const int*)  d_in[3];
    const float* Wg    = (const float*)d_in[4];
    const float* bias  = (const float*)d_in[5];
    const float* gamma = (const float*)d_in[6];
    const float* beta  = (const float*)d_in[7];

    const int N = in_sizes[0] / D;
    const int E = in_sizes[2];
    const int L = in_sizes[5] / D;
    float* out = (float*)d_out;

    // workspace layout (all fp32)
    float* deg   = (float*)d_ws;
    float* dinv  = deg  + N;
    float* h     = dinv + N;
    float* agg   = h    + (size_t)N * D;
    float* act   = agg  + (size_t)N * D;
    float* stats = act  + (size_t)N * D;

    const int* srcI = ei;       // edge_index[0]
    const int* dstI = ei + E;   // edge_index[1]

    const int t = 256;
    deg_init_kernel <<<(N + t - 1) / t, t, 0, stream>>>(deg, N);
    deg_accum_kernel<<<(E + t - 1) / t, t, 0, stream>>>(dstI, ew, deg, E);
    dinv_kernel     <<<(N + t - 1) / t, t, 0, stream>>>(deg, dinv, N);

    const float* cur = x;
    for (int l = 0; l < L; ++l) {
        gemm_wmma_kernel<<<(N + TILE_M - 1) / TILE_M, 256, 0, stream>>>(
            cur, Wg + (size_t)l * D * D, h, N);
        selfloop_bias_kernel<<<(N * D + t - 1) / t, t, 0, stream>>>(
            h, dinv, bias + l * D, agg, N);
        edge_scatter_kernel<<<8192, 256, 0, stream>>>(
            srcI, dstI, ew, dinv, h, agg, E);
        zero_kernel<<<1, 2 * D, 0, stream>>>(stats, 2 * D);
        bn_stats_kernel<<<1024, D, 0, stream>>>(agg, stats, N);
        bn_apply_kernel<<<(N * D + t - 1) / t, t, 0, stream>>>(
            agg, stats, gamma + l * D, beta + l * D, act, N);
        cur = act;
    }

    zero_kernel<<<(out_size + t - 1) / t, t, 0, stream>>>(out, out_size);
    pool_kernel<<<((size_t)N * 32 + t - 1) / t, t, 0, stream>>>(cur, batch, out, N);
}